// TransformerModel_82772609728550
// MI455X (gfx1250) — compile-verified
//
#include <hip/hip_runtime.h>
#include <hip/hip_bf16.h>

typedef float v2f __attribute__((ext_vector_type(2)));
typedef float v8f __attribute__((ext_vector_type(8)));

#define B_   512
#define S_   150
#define L_   50
#define D_   128
#define H_   8
#define DH_  16
#define FF_  256
#define NL_  2
#define DT_  0.1f
#define SP_  160            // S padded to multiple of 16
#define M_   (S_ * B_)      // 76800 rows, multiple of 16

static __device__ __forceinline__ v8f wmma_f32(v2f a, v2f b, v8f c) {
    // D = A(16x4 f32) * B(4x16 f32) + C(16x16 f32)
    return __builtin_amdgcn_wmma_f32_16x16x4_f32(false, a, false, b, (short)0, c,
                                                 false, false);
}

// ---------------------------------------------------------------- embedding
// X[s*B+b, d] = inputs[b,s,0] * emb_w[d] + emb_b[d]
__global__ void embed_kernel(const float* __restrict__ inp,
                             const float* __restrict__ emb_w,
                             const float* __restrict__ emb_b,
                             float* __restrict__ X) {
    int row = blockIdx.x;           // row = s*B + b
    int d   = threadIdx.x;          // 0..127
    int s   = row / B_;
    int b   = row % B_;
    float p = inp[((size_t)b * S_ + s) * 4 + 0];
    X[(size_t)row * D_ + d] = p * emb_w[d] + emb_b[d];
}

// ---------------------------------------------------------------- generic GEMM
// C[M,N] = act(A[M,K] @ W[N,K]^T + bias[N]);  one 16x16 tile per wave,
// K accumulated with chained V_WMMA_F32_16X16X4_F32.
template <int RELU>
__global__ void gemm_wmma_kernel(const float* __restrict__ A,
                                 const float* __restrict__ W,
                                 const float* __restrict__ bias,
                                 float* __restrict__ C,
                                 int N, int K) {
    int wave = threadIdx.x >> 5;
    int lane = threadIdx.x & 31;
    int m0 = blockIdx.y * 16;
    int n0 = (blockIdx.x * 4 + wave) * 16;
    if (n0 >= N) return;                       // wave-uniform

    int hf   = lane >> 4;       // half-wave: 0 -> K {0,1}, 1 -> K {2,3}
    int id16 = lane & 15;
    int koff = hf * 2;

    const float* Arow = A + (size_t)(m0 + id16) * K;
    const float* Wrow = W + (size_t)(n0 + id16) * K;

    v8f acc = {};
    for (int k0 = 0; k0 < K; k0 += 4) {
        v2f a = *(const v2f*)(Arow + k0 + koff);   // A[m, k0+koff .. +1]
        v2f b = *(const v2f*)(Wrow + k0 + koff);   // B[k,n] = W[n,k]
        acc = wmma_f32(a, b, acc);
    }

    int col   = n0 + id16;
    int rbase = m0 + hf * 8;
    float bn  = bias[col];
    for (int r = 0; r < 8; ++r) {
        float v = acc[r] + bn;
        if (RELU) v = v > 0.f ? v : 0.f;
        C[(size_t)(rbase + r) * N + col] = v;
    }
}

// ---------------------------------------------------------------- attention
// One workgroup per (b,h): q,k,v tiles + padded 160x160 score board in LDS.
__global__ void attn_kernel(const float* __restrict__ QKV,  // [S, B, 3D]
                            float* __restrict__ O) {        // [S, B, D]
    extern __shared__ float smem[];
    float* sQ = smem;                    // SP_*DH_
    float* sK = sQ + SP_ * DH_;
    float* sV = sK + SP_ * DH_;
    float* sS = sV + SP_ * DH_;          // SP_*SP_

    int bh  = blockIdx.x;
    int b   = bh / H_;
    int h   = bh % H_;
    int tid = threadIdx.x;

    for (int i = tid; i < SP_ * DH_; i += blockDim.x) {
        int srow = i / DH_, d = i % DH_;
        float q = 0.f, k = 0.f, v = 0.f;
        if (srow < S_) {
            size_t base = ((size_t)srow * B_ + b) * (3 * D_) + h * DH_ + d;
            q = QKV[base];
            k = QKV[base + D_];
            v = QKV[base + 2 * D_];
        }
        sQ[i] = q; sK[i] = k; sV[i] = v;
    }
    __syncthreads();

    int wave = tid >> 5, lane = tid & 31;
    int hf = lane >> 4, id16 = lane & 15, koff = hf * 2;
    const int NT = SP_ / 16;             // 10 tiles

    // scores = (Q K^T) * 1/sqrt(16); only lower-triangular tile pairs needed
    for (int t = wave; t < NT * NT; t += 8) {
        int mt = t / NT, nt = t % NT;
        if (nt > mt) continue;
        v8f acc = {};
        const float* qr = sQ + (mt * 16 + id16) * DH_;
        const float* kr = sK + (nt * 16 + id16) * DH_;
        for (int k0 = 0; k0 < DH_; k0 += 4) {
            v2f a, bb;
            a.x  = qr[k0 + koff];     a.y  = qr[k0 + koff + 1];
            bb.x = kr[k0 + koff];     bb.y = kr[k0 + koff + 1];
            acc = wmma_f32(a, bb, acc);
        }
        int rbase = mt * 16 + hf * 8;
        int col   = nt * 16 + id16;
        for (int r = 0; r < 8; ++r)
            sS[(rbase + r) * SP_ + col] = acc[r] * 0.25f;
    }
    __syncthreads();

    // causal row softmax; zero cols > row so the padded K-loop below is exact
    if (tid < S_) {
        float* row = sS + tid * SP_;
        float m = -1e30f;
        for (int c = 0; c <= tid; ++c) m = fmaxf(m, row[c]);
        float sum = 0.f;
        for (int c = 0; c <= tid; ++c) { float e = __expf(row[c] - m); row[c] = e; sum += e; }
        float inv = 1.f / sum;
        for (int c = 0; c <= tid; ++c) row[c] *= inv;
        for (int c = tid + 1; c < SP_; ++c) row[c] = 0.f;
    }
    __syncthreads();

    // O = P @ V  (K = SP_, padded rows of P are zero)
    for (int mt = wave; mt < NT; mt += 8) {
        v8f acc = {};
        const float* arow = sS + (mt * 16 + id16) * SP_;
        for (int k0 = 0; k0 < SP_; k0 += 4) {
            v2f a, bb;
            a.x  = arow[k0 + koff];
            a.y  = arow[k0 + koff + 1];
            bb.x = sV[(k0 + koff) * DH_ + id16];
            bb.y = sV[(k0 + koff + 1) * DH_ + id16];
            acc = wmma_f32(a, bb, acc);
        }
        int rbase = mt * 16 + hf * 8;
        for (int r = 0; r < 8; ++r) {
            int srow = rbase + r;
            if (srow < S_)
                O[((size_t)srow * B_ + b) * D_ + h * DH_ + id16] = acc[r];
        }
    }
}

// ---------------------------------------------------------------- residual+LN
// X[row,:] = LN(X[row,:] + R[row,:]) * g + beta ; one wave per 128-wide row
__global__ void ln_residual_kernel(float* __restrict__ X,
                                   const float* __restrict__ R,
                                   const float* __restrict__ g,
                                   const float* __restrict__ beta) {
    int row  = blockIdx.x * 8 + (threadIdx.x >> 5);
    int lane = threadIdx.x & 31;
    float* xr       = X + (size_t)row * D_;
    const float* rr = R + (size_t)row * D_;
    float vals[4];
    float s = 0.f;
    for (int i = 0; i < 4; ++i) {
        float v = xr[lane + 32 * i] + rr[lane + 32 * i];
        vals[i] = v; s += v;
    }
    for (int m = 16; m >= 1; m >>= 1) s += __shfl_xor(s, m, 32);
    float mu = s * (1.f / D_);
    float vs = 0.f;
    for (int i = 0; i < 4; ++i) { float d = vals[i] - mu; vs += d * d; }
    for (int m = 16; m >= 1; m >>= 1) vs += __shfl_xor(vs, m, 32);
    float inv = rsqrtf(vs * (1.f / D_) + 1e-5f);
    for (int i = 0; i < 4; ++i) {
        int d = lane + 32 * i;
        xr[d] = (vals[i] - mu) * inv * g[d] + beta[d];
    }
}

// ---------------------------------------------------------------- decoder dot
// dvec[b] = enc[b,:] . dec_w + dec_b   (enc = X[S-1])
__global__ void dec_kernel(const float* __restrict__ X,
                           const float* __restrict__ dec_w,
                           const float* __restrict__ dec_b,
                           float* __restrict__ dvec) {
    int b    = blockIdx.x * 8 + (threadIdx.x >> 5);
    int lane = threadIdx.x & 31;
    const float* enc = X + ((size_t)(S_ - 1) * B_ + b) * D_;
    float s = 0.f;
    for (int i = 0; i < 4; ++i) s += enc[lane + 32 * i] * dec_w[lane + 32 * i];
    for (int m = 16; m >= 1; m >>= 1) s += __shfl_xor(s, m, 32);
    if (lane == 0) dvec[b] = s + dec_b[0];
}

// ---------------------------------------------------------------- IDM scan
// One block of B_ threads; v0[-1] broadcast via LDS (reference quirk).
__global__ void idm_kernel(const float* __restrict__ inp,
                           const float* __restrict__ his,
                           const float* s0p, const float* Tp, const float* ap,
                           const float* bp, const float* vdp,
                           float* __restrict__ yidm) {
    __shared__ float v0s[B_];
    int b = threadIdx.x;
    float s0 = *s0p, T = *Tp, a = *ap, bpar = *bp, vd = *vdp;
    float sq = 2.f * sqrtf(a * bpar);

    const float* last = inp + ((size_t)b * S_ + (S_ - 1)) * 4;
    float y = last[0], v = last[1], s = last[2], dv = last[3];
    float sx  = s0 + fmaxf(0.f, v * T + v * dv / sq);
    float vr  = v / vd; float vr4 = vr * vr * vr * vr;
    float rat = sx / s;
    float af  = a * (1.f - vr4 - rat * rat);
    float v0  = fmaxf(v + af * DT_, 0.f);
    v0s[b] = v0;
    __syncthreads();
    float y0 = y + v0s[B_ - 1] * DT_;     // reference uses v0 of LAST batch row
    yidm[(size_t)b * L_] = y0;

    float vj = v0, yj = y0;
    for (int j = 0; j < L_ - 1; ++j) {
        float ly  = his[((size_t)b * L_ + j) * 2 + 0];
        float lv  = his[((size_t)b * L_ + j) * 2 + 1];
        float dvj = lv - vj;
        float sj  = ly - yj;
        float sxi = s0 + fmaxf(0.f, vj * T + vj * dvj / sq);
        float r   = vj / vd; float r4 = r * r * r * r;
        float rr  = sxi / sj;
        float acc = a * (1.f - r4 - rr * rr);
        float v2  = vj + acc * DT_;
        v2 = (v2 <= 0.f) ? 0.f : v2;
        float y2 = yj + v2 * DT_;
        yidm[(size_t)b * L_ + 1 + j] = y2;
        vj = v2; yj = y2;
    }
}

// ---------------------------------------------------------------- fusion
__global__ void fusion_kernel(const float* __restrict__ inp,
                              const float* __restrict__ dvec,
                              const float* __restrict__ yidm,
                              const float* __restrict__ fus_w,
                              const float* __restrict__ fus_b,
                              float* __restrict__ out) {
    int t = blockIdx.x * blockDim.x + threadIdx.x;
    if (t >= B_ * L_) return;
    int b = t / L_, l = t % L_;
    float last = inp[((size_t)b * S_ + (S_ - 1)) * 4];
    float prev = inp[((size_t)b * S_ + (S_ - 1 - L_)) * 4];
    float dvh  = (last - prev) / (float)L_;
    float hist = last + dvh * (float)(l + 1);
    out[t] = fus_w[0] * dvec[b] + fus_w[1] * hist + fus_w[2] * yidm[t] + fus_b[0];
}

// ---------------------------------------------------------------- launch
extern "C" void kernel_launch(void* const* d_in, const int* in_sizes, int n_in,
                              void* d_out, int out_size, void* d_ws, size_t ws_size,
                              hipStream_t stream) {
    const float* inputs = (const float*)d_in[0];
    const float* his    = (const float*)d_in[1];
    const float* s0p    = (const float*)d_in[2];
    const float* Tp     = (const float*)d_in[3];
    const float* ap     = (const float*)d_in[4];
    const float* bp     = (const float*)d_in[5];
    const float* vdp    = (const float*)d_in[6];
    const float* emb_w  = (const float*)d_in[7];
    const float* emb_b  = (const float*)d_in[8];
    const float* ipw    = (const float*)d_in[9];
    const float* ipb    = (const float*)d_in[10];
    const float* opw    = (const float*)d_in[11];
    const float* opb    = (const float*)d_in[12];
    const float* ln1g   = (const float*)d_in[13];
    const float* ln1b   = (const float*)d_in[14];
    const float* l1w    = (const float*)d_in[15];
    const float* l1b    = (const float*)d_in[16];
    const float* l2w    = (const float*)d_in[17];
    const float* l2b    = (const float*)d_in[18];
    const float* ln2g   = (const float*)d_in[19];
    const float* ln2b   = (const float*)d_in[20];
    const float* dec_w  = (const float*)d_in[21];
    const float* dec_b  = (const float*)d_in[22];
    const float* fus_w  = (const float*)d_in[23];
    const float* fus_b  = (const float*)d_in[24];
    float* out = (float*)d_out;

    // workspace layout (floats)
    float* ws   = (float*)d_ws;
    float* X    = ws;                               // M_ * D_
    float* QKV  = X   + (size_t)M_ * D_;            // M_ * 3D_ (reused below)
    float* O2   = QKV;                              // M_ * D_   (after attention)
    float* FFb  = QKV + (size_t)M_ * D_;            // M_ * FF_
    float* Obuf = QKV + (size_t)M_ * 3 * D_;        // M_ * D_
    float* dvec = Obuf + (size_t)M_ * D_;           // B_
    float* yidm = dvec + B_;                        // B_ * L_

    const int mtiles = M_ / 16;                     // 4800
    const size_t attn_lds = (size_t)(3 * SP_ * DH_ + SP_ * SP_) * sizeof(float);

    embed_kernel<<<M_, D_, 0, stream>>>(inputs, emb_w, emb_b, X);

    for (int l = 0; l < NL_; ++l) {
        // QKV = X @ ipw^T + ipb        (N = 3D = 384, K = 128)
        gemm_wmma_kernel<0><<<dim3(384 / 64, mtiles), 128, 0, stream>>>(
            X, ipw + (size_t)l * 3 * D_ * D_, ipb + (size_t)l * 3 * D_, QKV, 3 * D_, D_);

        // attention per (b, h)
        attn_kernel<<<B_ * H_, 256, attn_lds, stream>>>(QKV, Obuf);

        // O2 = Obuf @ opw^T + opb      (N = 128, K = 128); QKV region is free now
        gemm_wmma_kernel<0><<<dim3(128 / 64, mtiles), 128, 0, stream>>>(
            Obuf, opw + (size_t)l * D_ * D_, opb + (size_t)l * D_, O2, D_, D_);

        // X = LN(X + O2)
        ln_residual_kernel<<<M_ / 8, 256, 0, stream>>>(X, O2, ln1g + l * D_, ln1b + l * D_);

        // FFb = relu(X @ l1w^T + l1b)  (N = 256, K = 128)
        gemm_wmma_kernel<1><<<dim3(256 / 64, mtiles), 128, 0, stream>>>(
            X, l1w + (size_t)l * FF_ * D_, l1b + (size_t)l * FF_, FFb, FF_, D_);

        // O2 = FFb @ l2w^T + l2b       (N = 128, K = 256)
        gemm_wmma_kernel<0><<<dim3(128 / 64, mtiles), 128, 0, stream>>>(
            FFb, l2w + (size_t)l * D_ * FF_, l2b + (size_t)l * D_, O2, D_, FF_);

        // X = LN(X + O2)
        ln_residual_kernel<<<M_ / 8, 256, 0, stream>>>(X, O2, ln2g + l * D_, ln2b + l * D_);
    }

    dec_kernel<<<B_ / 8, 256, 0, stream>>>(X, dec_w, dec_b, dvec);
    idm_kernel<<<1, B_, 0, stream>>>(inputs, his, s0p, Tp, ap, bp, vdp, yidm);
    fusion_kernel<<<(B_ * L_ + 255) / 256, 256, 0, stream>>>(inputs, dvec, yidm,
                                                             fus_w, fus_b, out);
    (void)in_sizes; (void)n_in; (void)out_size; (void)ws_size;
}